// LRPlusL1Refiner_33715493274323
// MI455X (gfx1250) — compile-verified
//
#include <hip/hip_runtime.h>

typedef float v2f __attribute__((ext_vector_type(2)));
typedef float v8f __attribute__((ext_vector_type(8)));

#define N_TREES       500
#define LEAVES        64
#define TOTALK        32000            // 500 * 64
#define BATCH         2048
#define ROWS_PER_TILE 16
#define ROW_TILES     (BATCH / ROWS_PER_TILE)   // 128
#define KSPLIT        10
#define KCHUNK        (TOTALK / KSPLIT)         // 3200
#define TILE_K        128
#define STAGES        (KCHUNK / TILE_K)         // 25
#define LDS_STRIDE    132                       // 128 + 4 pad (conflict-free)

// Kernel 0: fold tree weights into leaf weights: w_eff[k] = leaf[k] * tree[k/64]
__global__ __launch_bounds__(256)
void gbm_fuse_weights(const float* __restrict__ lw, const float* __restrict__ tw,
                      float* __restrict__ weff) {
    int i = blockIdx.x * blockDim.x + threadIdx.x;
    if (i < TOTALK) weff[i] = lw[i] * tw[i / LEAVES];
}

// Kernel 1: split-K partial matvec via V_WMMA_F32_16X16X4_F32.
// Block = 256 threads (8 waves), computes 16 rows over a 3200-column K chunk.
__global__ __launch_bounds__(256)
void gbm_matvec_wmma(const float* __restrict__ x, const float* __restrict__ weff,
                     float* __restrict__ partial) {
    __shared__ float ldsA[ROWS_PER_TILE * LDS_STRIDE];   // 16 x 128 tile, padded
    __shared__ float red[8 * ROWS_PER_TILE];             // per-wave row partials

    const int tid  = threadIdx.x;
    const int wave = tid >> 5;
    const int lane = tid & 31;
    const int half = lane >> 4;     // 0: lanes 0-15, 1: lanes 16-31
    const int m    = lane & 15;     // matrix row within the tile

    const int rowBase  = blockIdx.x * ROWS_PER_TILE;
    const int colStart = blockIdx.y * KCHUNK;

    v8f c = {};

    for (int s = 0; s < STAGES; ++s) {
        const int tileCol = colStart + s * TILE_K;

        // --- Stage 16x128 f32 tile into LDS (coalesced float4 global loads).
        // 512 float4s total; each thread handles 2.
        #pragma unroll
        for (int rep = 0; rep < 2; ++rep) {
            const int f  = tid + rep * 256;   // float4 index 0..511
            const int r  = f >> 5;            // tile row (32 float4 per row)
            const int cv = f & 31;            // float4 column within row
            const float4 v = *(const float4*)(x + (size_t)(rowBase + r) * TOTALK
                                                + tileCol + cv * 4);
            float* dst = &ldsA[r * LDS_STRIDE + cv * 4];
            dst[0] = v.x; dst[1] = v.y; dst[2] = v.z; dst[3] = v.w;
        }
        __syncthreads();

        // --- Each wave consumes its 16 columns of the tile: 4 WMMAs of K=4.
        const int waveCol = wave * 16;
        #pragma unroll
        for (int kk = 0; kk < 4; ++kk) {
            const int c0 = waveCol + kk * 4;
            // A fragment (16x4 f32): lanes 0-15 hold K={c0,c0+1} for row m,
            // lanes 16-31 hold K={c0+2,c0+3} (ISA 32-bit A layout).
            v2f a;
            const float* ap = &ldsA[m * LDS_STRIDE + c0 + 2 * half];
            a.x = ap[0];
            a.y = ap[1];
            // B fragment (4x16 f32): broadcast w_eff over all 16 N columns,
            // same per-lane K mapping as A.
            const int g = tileCol + c0 + 2 * half;
            v2f b;
            b.x = weff[g];
            b.y = weff[g + 1];
            // D = A*B + C  (8 args: neg_a, A, neg_b, B, c_mod, C, reuse_a, reuse_b)
            c = __builtin_amdgcn_wmma_f32_16x16x4_f32(
                    false, a, false, b, (short)0, c, false, false);
        }
        __syncthreads();
    }

    // Every N column of D holds the same value (B was broadcast).
    // D layout: VGPR p -> row p (lanes 0-15) / row p+8 (lanes 16-31).
    if (lane == 0 || lane == 16) {
        #pragma unroll
        for (int p = 0; p < 8; ++p)
            red[wave * 16 + half * 8 + p] = c[p];
    }
    __syncthreads();

    // Cross-wave reduce (deterministic, fixed order), write split-K partial.
    if (tid < ROWS_PER_TILE) {
        float acc = 0.f;
        #pragma unroll
        for (int w = 0; w < 8; ++w) acc += red[w * 16 + tid];
        partial[(size_t)blockIdx.y * BATCH + rowBase + tid] = acc;
    }
}

// Kernel 2: sum split-K partials + base score.
__global__ __launch_bounds__(256)
void gbm_reduce(const float* __restrict__ partial, const float* __restrict__ base,
                float* __restrict__ out) {
    const int i = blockIdx.x * blockDim.x + threadIdx.x;
    float acc = base[0];
    #pragma unroll
    for (int s = 0; s < KSPLIT; ++s) acc += partial[(size_t)s * BATCH + i];
    out[i] = acc;
}

extern "C" void kernel_launch(void* const* d_in, const int* in_sizes, int n_in,
                              void* d_out, int out_size, void* d_ws, size_t ws_size,
                              hipStream_t stream) {
    const float* x    = (const float*)d_in[0];   // [2048, 32000]
    const float* lw   = (const float*)d_in[1];   // [32000]
    const float* tw   = (const float*)d_in[2];   // [500]
    const float* base = (const float*)d_in[3];   // [1]
    float* out = (float*)d_out;                  // [2048]

    // Workspace layout: [0, 32000) w_eff ; [32000, 32000 + 10*2048) partials
    float* weff    = (float*)d_ws;
    float* partial = weff + TOTALK;

    gbm_fuse_weights<<<(TOTALK + 255) / 256, 256, 0, stream>>>(lw, tw, weff);
    gbm_matvec_wmma<<<dim3(ROW_TILES, KSPLIT), 256, 0, stream>>>(x, weff, partial);
    gbm_reduce<<<BATCH / 256, 256, 0, stream>>>(partial, base, out);
}